// WaveletHFLoss_64372969832982
// MI455X (gfx1250) — compile-verified
//
#include <hip/hip_runtime.h>

// Wavelet HF loss, algebraically reduced:
//   loss = (1/N) * sum over 2x2 blocks of (sum d^2 - 0.25*(sum d)^2),  d = sr - hr
// Memory-bound: 201 MB single-use read -> NT b128 loads; WMMA f32 16x16x4
// used for the deterministic intra-wave reduction.

typedef __attribute__((ext_vector_type(2))) float v2f;
typedef __attribute__((ext_vector_type(4))) float v4f;
typedef __attribute__((ext_vector_type(8))) float v8f;

namespace {
constexpr unsigned kW           = 1024u;
constexpr unsigned kRows        = 8u * 3u * 1024u;            // 24576 contiguous rows
constexpr unsigned kRowPairs    = kRows / 2u;                 // 12288
constexpr unsigned kQuadsPerRow = kW / 4u;                    // 256
constexpr unsigned kUnits       = kRowPairs * kQuadsPerRow;   // 3,145,728
constexpr float    kInvN        = 1.0f / 25165824.0f;         // 8*3*1024*1024
constexpr int      kBlock       = 256;                        // 8 waves (wave32)
constexpr int      kMaxBlocks   = 2048;
}

// Wave32 sum via V_WMMA_F32_16X16X4_F32.
// A layout (16x4 f32): VGPR0 = {lanes0-15: A[m,0], lanes16-31: A[m,2]},
//                      VGPR1 = {lanes0-15: A[m,1], lanes16-31: A[m,3]}.
// With a = {p, 0} per lane and B = ones: D[m,n] = p_m + p_{m+16}.
// Lane n<16 holds D[0..7, n], lane n>=16 holds D[8..15, n-16], so
// sum(8 D regs) + shfl_xor(16) = full wave sum in every lane.
__device__ __forceinline__ float wave_sum_wmma(float p) {
  v2f a; a[0] = p;    a[1] = 0.0f;
  v2f b; b[0] = 1.0f; b[1] = 1.0f;
  v8f c = {};
  c = __builtin_amdgcn_wmma_f32_16x16x4_f32(false, a, false, b, (short)0, c,
                                            false, false);
  float s = c[0] + c[1] + c[2] + c[3] + c[4] + c[5] + c[6] + c[7];
  s += __shfl_xor(s, 16, 32);
  return s;
}

// Deterministic block reduction: per-wave WMMA sum -> LDS -> thread 0
// sequential sum -> *dst = total * scale.
__device__ __forceinline__ void block_reduce_store(float p, float* dst,
                                                   float scale) {
  __shared__ float wsum[kBlock / 32];
  float s = wave_sum_wmma(p);
  const int lane = threadIdx.x & 31;
  const int wave = threadIdx.x >> 5;
  if (lane == 0) wsum[wave] = s;
  __syncthreads();
  if (threadIdx.x == 0) {
    float t = 0.0f;
    const int nw = blockDim.x >> 5;
    for (int i = 0; i < nw; ++i) t += wsum[i];
    *dst = t * scale;
  }
}

__global__ __launch_bounds__(kBlock) void hf_loss_partial(
    const float* __restrict__ sr, const float* __restrict__ hr,
    float* __restrict__ partials) {
  const unsigned stride = gridDim.x * blockDim.x;
  float acc = 0.0f;

  for (unsigned u = blockIdx.x * blockDim.x + threadIdx.x; u < kUnits;
       u += stride) {
    const unsigned rp  = u >> 8;          // row pair index
    const unsigned c4  = u & 255u;        // 4-column group within the row
    const unsigned top = rp * (2u * kW) + c4 * 4u;

    // Single-use streaming data: non-temporal 128-bit loads.
    const v4f st = __builtin_nontemporal_load((const v4f*)(sr + top));
    const v4f sb = __builtin_nontemporal_load((const v4f*)(sr + top + kW));
    const v4f ht = __builtin_nontemporal_load((const v4f*)(hr + top));
    const v4f hb = __builtin_nontemporal_load((const v4f*)(hr + top + kW));

    const float d0 = st[0] - ht[0], d1 = st[1] - ht[1];
    const float d2 = st[2] - ht[2], d3 = st[3] - ht[3];
    const float e0 = sb[0] - hb[0], e1 = sb[1] - hb[1];
    const float e2 = sb[2] - hb[2], e3 = sb[3] - hb[3];

    // Two 2x2 blocks per unit: Parseval form  sum(d^2) - 0.25*(sum d)^2
    const float t0 = (d0 + d1) + (e0 + e1);
    const float t1 = (d2 + d3) + (e2 + e3);
    acc += d0 * d0 + d1 * d1 + e0 * e0 + e1 * e1 - 0.25f * t0 * t0;
    acc += d2 * d2 + d3 * d3 + e2 * e2 + e3 * e3 - 0.25f * t1 * t1;
  }

  block_reduce_store(acc, &partials[blockIdx.x], 1.0f);
}

__global__ __launch_bounds__(kBlock) void hf_loss_finalize(
    const float* __restrict__ partials, int nb, float* __restrict__ out) {
  float p = 0.0f;
  for (int i = threadIdx.x; i < nb; i += blockDim.x) p += partials[i];
  block_reduce_store(p, out, kInvN);
}

extern "C" void kernel_launch(void* const* d_in, const int* in_sizes, int n_in,
                              void* d_out, int out_size, void* d_ws,
                              size_t ws_size, hipStream_t stream) {
  (void)in_sizes; (void)n_in; (void)out_size;
  const float* sr = (const float*)d_in[0];
  const float* hr = (const float*)d_in[1];
  float* out      = (float*)d_out;
  float* partials = (float*)d_ws;

  int nb = kMaxBlocks;
  const size_t cap = ws_size / sizeof(float);
  if ((size_t)nb > cap) nb = (int)cap;
  if (nb < 1) nb = 1;

  hf_loss_partial<<<nb, kBlock, 0, stream>>>(sr, hr, partials);
  hf_loss_finalize<<<1, kBlock, 0, stream>>>(partials, nb, out);
}